// LossEvaluator_90082644066695
// MI455X (gfx1250) — compile-verified
//
#include <hip/hip_runtime.h>

// MI455X / gfx1250, wave32. One wave per batch column; each lane owns 2 sample
// trajectories. Cross-sample mean (baseline) done with exact-f32 WMMA
// (V_WMMA_F32_16X16X4_F32) lane reductions; next-step inputs prefetched
// (global_prefetch_b8) to hide HBM latency across the serial scan.

typedef __attribute__((ext_vector_type(2))) float v2f;
typedef __attribute__((ext_vector_type(8))) float v8f;

#define SEQ_LEN 128
#define NCUR 4
#define NSAMP 64
#define NBATCH 128
#define SB (NSAMP * NBATCH)
#define LEV 50.0f

__device__ __forceinline__ v8f wmma16x16x4(v2f a, v2f b, v8f c) {
  // 8 args: (neg_a, A, neg_b, B, c_mod, C, reuse_a, reuse_b)
  return __builtin_amdgcn_wmma_f32_16x16x4_f32(false, a, false, b, (short)0, c,
                                               false, false);
}

// Sum a per-lane f32 value over all 32 lanes, result broadcast to every lane.
// Exact f32 math: B is all-ones so only the documented A and C/D layouts
// matter.  Stage 1: D[m,n] = v_m + v_{m+16}.  Stage 2: accumulate the 16
// pair-sums (rows of D land in lanes per the documented C/D layout).
__device__ __forceinline__ float wave_sum32(float x) {
  v2f a;   a[0] = x;   a[1] = 0.0f;
  v2f one; one[0] = 1.0f; one[1] = 1.0f;
  v8f zero = {0.f, 0.f, 0.f, 0.f, 0.f, 0.f, 0.f, 0.f};
  v8f d = wmma16x16x4(a, one, zero);      // d[r] lanes0-15 = p_r, lanes16-31 = p_{r+8}
  v8f acc = zero;
  v2f t;
  t[0] = d[0]; t[1] = d[1]; acc = wmma16x16x4(t, one, acc);  // +p0+p1+p8+p9
  t[0] = d[2]; t[1] = d[3]; acc = wmma16x16x4(t, one, acc);  // +p2+p3+p10+p11
  t[0] = d[4]; t[1] = d[5]; acc = wmma16x16x4(t, one, acc);  // +p4+p5+p12+p13
  t[0] = d[6]; t[1] = d[7]; acc = wmma16x16x4(t, one, acc);  // +p6+p7+p14+p15
  return acc[0];                           // every lane: sum of all 32 inputs
}

__global__ __launch_bounds__(128, 1) void trade_scan_kernel(
    const int* __restrict__ g_smp,   // (T,3,4,64,128) int32
    const float* __restrict__ g_frac,// (T,4,64,128)
    const float* __restrict__ g_xlp, // (T,3,4,64,128)
    const float* __restrict__ g_zlp, // (T,64,128)
    const float* __restrict__ g_rt,  // (T,2,4,128)
    const float* __restrict__ g_bc,  // (T,4,128)
    float* __restrict__ g_out)       // (64,128)
{
  const int lane = threadIdx.x & 31;
  const int wv = threadIdx.x >> 5;
  const int b = blockIdx.x * 4 + wv;   // one wave per batch column

  int sbo[2];
  sbo[0] = lane * NBATCH + b;          // sample s = lane
  sbo[1] = (lane + 32) * NBATCH + b;   // sample s = lane + 32

  // Per-trajectory state (all registers).
  int pstate[2][NCUR], ptype[2][NCUR];
  float cumz[2], tmarg[2], lossv[2];
  float opsz[2][NCUR], orate[2][NCUR], olp[2][NCUR];
  float cexec[2][NCUR], cclose[2][NCUR], otlp[2][NCUR];
#pragma unroll
  for (int p = 0; p < 2; ++p) {
    cumz[p] = 0.f; tmarg[p] = 1.f; lossv[p] = 0.f;
#pragma unroll
    for (int j = 0; j < NCUR; ++j) {
      pstate[p][j] = 0; ptype[p][j] = 0;
      opsz[p][j] = 0.f; orate[p][j] = 0.f; olp[p][j] = 0.f;
      cexec[p][j] = 0.f; cclose[p][j] = 0.f; otlp[p][j] = 0.f;
    }
  }

#pragma unroll 1
  for (int t = 0; t < SEQ_LEN; ++t) {
    const int* smp_t = g_smp + t * 3 * NCUR * SB;
    const float* xlp_t = g_xlp + t * 3 * NCUR * SB;
    const float* fr_t = g_frac + t * NCUR * SB;
    const float* z_t = g_zlp + t * SB;
    const float* r_t = g_rt + t * 2 * NCUR * NBATCH;
    const float* bc_t = g_bc + t * NCUR * NBATCH;

    // --- Prefetch all of step t+1's per-thread data (hides HBM latency in the
    // serial scan; prefetches don't touch LOADcnt). Clamped => uniform code.
    {
      const int tn = (t + 1 < SEQ_LEN) ? (t + 1) : t;
      const int* smp_n = g_smp + tn * 3 * NCUR * SB;
      const float* xlp_n = g_xlp + tn * 3 * NCUR * SB;
      const float* fr_n = g_frac + tn * NCUR * SB;
      const float* z_n = g_zlp + tn * SB;
#pragma unroll
      for (int p = 0; p < 2; ++p) {
#pragma unroll
        for (int kj = 0; kj < 12; ++kj) {
          __builtin_prefetch(smp_n + kj * SB + sbo[p]);
          __builtin_prefetch(xlp_n + kj * SB + sbo[p]);
        }
#pragma unroll
        for (int j = 0; j < NCUR; ++j) __builtin_prefetch(fr_n + j * SB + sbo[p]);
        __builtin_prefetch(z_n + sbo[p]);
      }
    }

    // --- Wave-uniform per-batch inputs.
    float r0[NCUR], r1[NCUR], bc[NCUR], rbc[NCUR], rr0[NCUR], rr1[NCUR];
#pragma unroll
    for (int j = 0; j < NCUR; ++j) {
      r0[j] = r_t[j * NBATCH + b];
      r1[j] = r_t[NCUR * NBATCH + j * NBATCH + b];
      bc[j] = bc_t[j * NBATCH + b];
      rbc[j] = 1.0f / bc[j];
      rr0[j] = 1.0f / r0[j];
      rr1[j] = 1.0f / r1[j];
    }

    float pl[2][NCUR], costs[2][NCUR], mav[2][NCUR];
    bool om[2][NCUR], cm[2][NCUR];

#pragma unroll
    for (int p = 0; p < 2; ++p) {
      // Pass 1: P&L / margin / closeout from ORIGINAL state.
      float plsum = 0.f, mu = 0.f;
#pragma unroll
      for (int j = 0; j < NCUR; ++j) {
        bool isopen = (pstate[p][j] == 1);
        float rcr = isopen ? ((ptype[p][j] == 0) ? rr0[j] : rr1[j]) : 1.0f;
        float bo = opsz[p][j] * rbc[j];
        float pj = LEV * bo * (1.0f - orate[p][j] * rcr);
        pl[p][j] = pj; plsum += pj; mu += bo;
      }
      float av = tmarg[p] + plsum;
      bool closeout = av < 0.5f * mu;
      cumz[p] += z_t[sbo[p]];
      float sumce = cexec[p][0] + cexec[p][1] + cexec[p][2] + cexec[p][3];
      float run = 0.f;

      // Pass 2: masks, state toggles, logprob accumulators.
#pragma unroll
      for (int j = 0; j < NCUR; ++j) {
        int so = smp_t[(0 * NCUR + j) * SB + sbo[p]];
        int sc = smp_t[(1 * NCUR + j) * SB + sbo[p]];
        int spt = smp_t[(2 * NCUR + j) * SB + sbo[p]];
        float x0 = xlp_t[(0 * NCUR + j) * SB + sbo[p]];
        float x1 = xlp_t[(1 * NCUR + j) * SB + sbo[p]];
        float x2 = xlp_t[(2 * NCUR + j) * SB + sbo[p]];
        bool isopen = (pstate[p][j] == 1);
        bool isclosed = !isopen;
        bool o = isclosed && (!closeout) && (so == 1);
        bool c = isopen && (closeout || (sc == 1));
        om[p][j] = o; cm[p][j] = c;
        pstate[p][j] = (o || c) ? (pstate[p][j] ^ 1) : pstate[p][j];
        float olpv = ((!closeout) && isclosed) ? x0 : 0.f;
        float clpv = ((!closeout) && isopen) ? x1 : 0.f;
        float el = olpv + clpv;
        run += el;
        mav[p][j] = sumce + run;           // old cum_exec sum + inclusive cumsum
        cexec[p][j] += el;
        cclose[p][j] += clpv;
        otlp[p][j] = o ? x2 : otlp[p][j];
        int newt = o ? spt : ptype[p][j];
        ptype[p][j] = newt;
        orate[p][j] = (o && newt == 0) ? r0[j]
                     : ((o && newt == 1) ? r1[j] : orate[p][j]);
        costs[p][j] = c ? pl[p][j] : 0.f;
      }
    }

    // --- Baselines: mean over 64 samples per currency, via exact-f32 WMMA
    // reductions (EXEC is all-ones: grid is fully populated, code branchless).
    float base[NCUR];
#pragma unroll
    for (int j = 0; j < NCUR; ++j) {
      float csum = costs[0][j] + costs[1][j];          // pre-add the 2 samples
      base[j] = wave_sum32(csum) * (1.0f / 64.0f);
    }

    // --- Pass 3: sequential per-currency loss / margin update (order matters:
    // margin_used_j sees opsz/tmarg updates from previous j iterations).
#pragma unroll
    for (int j = 0; j < NCUR; ++j) {
#pragma unroll
      for (int p = 0; p < 2; ++p) {
        float muj = opsz[p][0] * rbc[0] + opsz[p][1] * rbc[1] +
                    opsz[p][2] * rbc[2] + opsz[p][3] * rbc[3];
        float mavail = tmarg[p] - muj;
        float nps = fr_t[j * SB + sbo[p]] * mavail * bc[j];
        bool o = om[p][j];
        opsz[p][j] = o ? nps : opsz[p][j];
        olp[p][j] = o ? mav[p][j] : olp[p][j];
        bool c = cm[p][j];
        float cost = costs[p][j];
        float clogp = cumz[p] + olp[p][j] + otlp[p][j] + cclose[p][j];
        lossv[p] += c ? (clogp * (cost - base[j]) + cost) : 0.f;
        opsz[p][j] = c ? 0.f : opsz[p][j];
        tmarg[p] += c ? pl[p][j] : 0.f;
      }
    }
#pragma unroll
    for (int p = 0; p < 2; ++p)
#pragma unroll
      for (int j = 0; j < NCUR; ++j)
        cclose[p][j] = cm[p][j] ? 0.f : cclose[p][j];
  }

  g_out[sbo[0]] = lossv[0];
  g_out[sbo[1]] = lossv[1];
}

extern "C" void kernel_launch(void* const* d_in, const int* in_sizes, int n_in,
                              void* d_out, int out_size, void* d_ws,
                              size_t ws_size, hipStream_t stream) {
  const int* samples = (const int*)d_in[0];
  const float* fractions = (const float*)d_in[1];
  const float* x_logprobs = (const float*)d_in[2];
  const float* z_logprobs = (const float*)d_in[3];
  const float* rates = (const float*)d_in[4];
  const float* base_cur_rates = (const float*)d_in[5];
  float* out = (float*)d_out;
  (void)in_sizes; (void)n_in; (void)out_size; (void)d_ws; (void)ws_size;

  dim3 grid(NBATCH / 4);   // 32 blocks
  dim3 block(128);         // 4 wave32s per block, one wave per batch column
  trade_scan_kernel<<<grid, block, 0, stream>>>(
      samples, fractions, x_logprobs, z_logprobs, rates, base_cur_rates, out);
}